// _GNN_24927990186433
// MI455X (gfx1250) — compile-verified
//
#include <hip/hip_runtime.h>
#include <cstddef>

namespace {

constexpr int kN  = 100000;   // nodes
constexpr int kE  = 1600000;  // edges
constexpr int kG  = 64;       // graphs
constexpr int kD0 = 128;      // input node dim
constexpr int kH  = 64;       // hidden

typedef float v2f __attribute__((ext_vector_type(2)));
typedef float v8f __attribute__((ext_vector_type(8)));

// ---------------- degree / norm ----------------
__global__ void k_deg(const int* __restrict__ dst, float* __restrict__ deg, int nE) {
  int e = blockIdx.x * blockDim.x + threadIdx.x;
  if (e < nE) atomicAdd(&deg[dst[e]], 1.0f);
}

__global__ void k_norm(float* __restrict__ deg, int nN) {
  int n = blockIdx.x * blockDim.x + threadIdx.x;
  if (n < nN) deg[n] = 1.0f / sqrtf(fmaxf(deg[n], 1.0f));
}

// ---------------- sparse hop: agg[dst] += h[src] * norm[src] ----------------
template <int D, bool SAN>
__global__ void k_scatter(const float* __restrict__ h, const float* __restrict__ norm,
                          const int* __restrict__ src, const int* __restrict__ dst,
                          float* __restrict__ agg, int nE) {
  constexpr int CPE = D / 4;  // float4 chunks per edge
  int tid = blockIdx.x * blockDim.x + threadIdx.x;
  if (tid >= nE * CPE) return;
  int e = tid / CPE;
  int c = tid - e * CPE;
  int s = src[e];
  int d = dst[e];
  float ns = norm[s];
  float4 v = *(const float4*)(h + (size_t)s * D + c * 4);
  if (SAN) {  // nan_to_num on raw input x
    v.x = (v.x != v.x) ? 0.f : v.x;
    v.y = (v.y != v.y) ? 0.f : v.y;
    v.z = (v.z != v.z) ? 0.f : v.z;
    v.w = (v.w != v.w) ? 0.f : v.w;
  }
  float* o = agg + (size_t)d * D + c * 4;
  atomicAdd(o + 0, v.x * ns);
  atomicAdd(o + 1, v.y * ns);
  atomicAdd(o + 2, v.z * ns);
  atomicAdd(o + 3, v.w * ns);
}

// ---------------- h[n] *= norm[n] ----------------
template <int D>
__global__ void k_scale(float* __restrict__ h, const float* __restrict__ norm, int nN) {
  constexpr int CPN = D / 4;
  int tid = blockIdx.x * blockDim.x + threadIdx.x;
  if (tid >= nN * CPN) return;
  int n = tid / CPN;
  float s = norm[n];
  float4* p = (float4*)h + tid;
  float4 v = *p;
  v.x *= s; v.y *= s; v.z *= s; v.w *= s;
  *p = v;
}

// ---------------- fp32 WMMA GEMM: out = relu([h0|h1|h2] @ W + b) ----------------
// One wave computes a 16x64 tile. A: 16x4 f32 (2 VGPR / lane), B: 4x16 (2 VGPR),
// C/D: 16x16 f32 (8 VGPR). Lanes 0-15 carry K+{0,1}, lanes 16-31 carry K+{2,3}.
template <int D, bool SAN>
__device__ __forceinline__ void tag_piece(const float* __restrict__ h,
                                          const float* __restrict__ Wp,
                                          int row0, int lm, int half, v8f acc[4]) {
  const float* arow = h + (size_t)(row0 + lm) * D + 2 * half;
  for (int k = 0; k < D; k += 4) {
    v2f a = *(const v2f*)(arow + k);
    if (SAN) {
      a.x = (a.x != a.x) ? 0.f : a.x;
      a.y = (a.y != a.y) ? 0.f : a.y;
    }
    const float* wk = Wp + (size_t)(k + 2 * half) * kH + lm;
#pragma unroll
    for (int nt = 0; nt < 4; ++nt) {
      v2f b;
      b.x = wk[nt * 16];       // W[k + 2*half][col]
      b.y = wk[kH + nt * 16];  // W[k + 2*half + 1][col]
      acc[nt] = __builtin_amdgcn_wmma_f32_16x16x4_f32(
          false, a, false, b, (short)0, acc[nt], false, false);
    }
  }
}

template <int D, bool SAN>
__global__ __launch_bounds__(256) void k_gemm(const float* __restrict__ h0,
                                              const float* __restrict__ h1,
                                              const float* __restrict__ h2,
                                              const float* __restrict__ W,
                                              const float* __restrict__ bias,
                                              float* __restrict__ out, int nN) {
  int wave = blockIdx.x * (blockDim.x >> 5) + (threadIdx.x >> 5);
  int row0 = wave * 16;
  if (row0 >= nN) return;  // wave-uniform: EXEC stays all-ones for WMMA
  int lane = threadIdx.x & 31;
  int half = lane >> 4;
  int lm   = lane & 15;

  v8f acc[4];
#pragma unroll
  for (int t = 0; t < 4; ++t)
#pragma unroll
    for (int i = 0; i < 8; ++i) acc[t][i] = 0.0f;

  tag_piece<D, SAN>(h0, W, row0, lm, half, acc);
  tag_piece<D, false>(h1, W + (size_t)D * kH, row0, lm, half, acc);
  tag_piece<D, false>(h2, W + (size_t)2 * D * kH, row0, lm, half, acc);

#pragma unroll
  for (int nt = 0; nt < 4; ++nt) {
    int col = nt * 16 + lm;
    float bcol = bias[col];
#pragma unroll
    for (int v = 0; v < 8; ++v) {
      int m = v + half * 8;  // D layout: vgpr v -> M=v (lanes 0-15) / M=v+8 (16-31)
      float val = acc[nt][v] + bcol;
      out[(size_t)(row0 + m) * kH + col] = fmaxf(val, 0.0f);
    }
  }
}

// ---------------- per-graph mean + classifier ----------------
__global__ void k_gsum(const float* __restrict__ h, const int* __restrict__ gid,
                       float* __restrict__ hsum, float* __restrict__ cnt, int nN) {
  int n = blockIdx.x;
  int f = threadIdx.x;
  if (n >= nN) return;
  int g = gid[n];
  atomicAdd(&hsum[g * kH + f], h[(size_t)n * kH + f]);
  if (f == 0) atomicAdd(&cnt[g], 1.0f);
}

__global__ void k_gout(const float* __restrict__ hsum, const float* __restrict__ cnt,
                       const float* __restrict__ Wc, const float* __restrict__ bc,
                       float* __restrict__ out) {
  int g = threadIdx.x;
  if (g >= kG) return;
  float c = fmaxf(cnt[g], 1.0f);
  float s = 0.0f;
  for (int f = 0; f < kH; ++f) s += (hsum[g * kH + f] / c) * Wc[f];
  out[g] = s + bc[0];
}

inline int cdiv(long a, long b) { return (int)((a + b - 1) / b); }

}  // namespace

extern "C" void kernel_launch(void* const* d_in, const int* in_sizes, int n_in,
                              void* d_out, int out_size, void* d_ws, size_t ws_size,
                              hipStream_t stream) {
  (void)in_sizes; (void)n_in; (void)out_size; (void)ws_size;
  const float* x  = (const float*)d_in[0];
  const float* W1 = (const float*)d_in[1];
  const float* b1 = (const float*)d_in[2];
  const float* W2 = (const float*)d_in[3];
  const float* b2 = (const float*)d_in[4];
  const float* W3 = (const float*)d_in[5];
  const float* b3 = (const float*)d_in[6];
  const float* Wc = (const float*)d_in[7];
  const float* bc = (const float*)d_in[8];
  const int* src  = (const int*)d_in[9];
  const int* dst  = (const int*)d_in[10];
  const int* gid  = (const int*)d_in[11];

  float* ws   = (float*)d_ws;
  float* norm = ws;                               // kN (kN % 4 == 0)
  float* hopA = norm + kN;                        // kN * kD0 capacity
  float* hopB = hopA + (size_t)kN * kD0;          // kN * kD0 capacity
  float* hX1  = hopB + (size_t)kN * kD0;          // kN * kH
  float* hX2  = hX1 + (size_t)kN * kH;            // kN * kH
  float* hsum = hX2 + (size_t)kN * kH;            // kG * kH
  float* cnt  = hsum + kG * kH;                   // kG

  const int B = 256;
  const int gemmBlocks = cdiv(kN / 16, 8);        // 8 waves per 256-thread block

  // degree -> norm = clip(deg,1)^-0.5
  hipMemsetAsync(norm, 0, kN * sizeof(float), stream);
  k_deg<<<cdiv(kE, B), B, 0, stream>>>(dst, norm, kE);
  k_norm<<<cdiv(kN, B), B, 0, stream>>>(norm, kN);

  // ---- layer 1 (D = 128) ----
  hipMemsetAsync(hopA, 0, (size_t)kN * kD0 * sizeof(float), stream);
  k_scatter<kD0, true><<<cdiv((long)kE * (kD0 / 4), B), B, 0, stream>>>(x, norm, src, dst, hopA, kE);
  k_scale<kD0><<<cdiv((long)kN * (kD0 / 4), B), B, 0, stream>>>(hopA, norm, kN);
  hipMemsetAsync(hopB, 0, (size_t)kN * kD0 * sizeof(float), stream);
  k_scatter<kD0, false><<<cdiv((long)kE * (kD0 / 4), B), B, 0, stream>>>(hopA, norm, src, dst, hopB, kE);
  k_scale<kD0><<<cdiv((long)kN * (kD0 / 4), B), B, 0, stream>>>(hopB, norm, kN);
  k_gemm<kD0, true><<<gemmBlocks, B, 0, stream>>>(x, hopA, hopB, W1, b1, hX1, kN);

  // ---- layer 2 (D = 64) ----
  hipMemsetAsync(hopA, 0, (size_t)kN * kH * sizeof(float), stream);
  k_scatter<kH, false><<<cdiv((long)kE * (kH / 4), B), B, 0, stream>>>(hX1, norm, src, dst, hopA, kE);
  k_scale<kH><<<cdiv((long)kN * (kH / 4), B), B, 0, stream>>>(hopA, norm, kN);
  hipMemsetAsync(hopB, 0, (size_t)kN * kH * sizeof(float), stream);
  k_scatter<kH, false><<<cdiv((long)kE * (kH / 4), B), B, 0, stream>>>(hopA, norm, src, dst, hopB, kE);
  k_scale<kH><<<cdiv((long)kN * (kH / 4), B), B, 0, stream>>>(hopB, norm, kN);
  k_gemm<kH, false><<<gemmBlocks, B, 0, stream>>>(hX1, hopA, hopB, W2, b2, hX2, kN);

  // ---- layer 3 (D = 64), output overwrites hX1 ----
  hipMemsetAsync(hopA, 0, (size_t)kN * kH * sizeof(float), stream);
  k_scatter<kH, false><<<cdiv((long)kE * (kH / 4), B), B, 0, stream>>>(hX2, norm, src, dst, hopA, kE);
  k_scale<kH><<<cdiv((long)kN * (kH / 4), B), B, 0, stream>>>(hopA, norm, kN);
  hipMemsetAsync(hopB, 0, (size_t)kN * kH * sizeof(float), stream);
  k_scatter<kH, false><<<cdiv((long)kE * (kH / 4), B), B, 0, stream>>>(hopA, norm, src, dst, hopB, kE);
  k_scale<kH><<<cdiv((long)kN * (kH / 4), B), B, 0, stream>>>(hopB, norm, kN);
  k_gemm<kH, false><<<gemmBlocks, B, 0, stream>>>(hX2, hopA, hopB, W3, b3, hX1, kN);

  // ---- readout: per-graph mean + linear ----
  hipMemsetAsync(hsum, 0, (size_t)(kG * kH + kG) * sizeof(float), stream);
  k_gsum<<<kN, kH, 0, stream>>>(hX1, gid, hsum, cnt, kN);
  k_gout<<<1, kH, 0, stream>>>(hsum, cnt, Wc, bc, (float*)d_out);
}